// Net_15307263443560
// MI455X (gfx1250) — compile-verified
//
#include <hip/hip_runtime.h>
#include <math.h>

// ---------------------------------------------------------------------------
// GNN forward for the reference Net (D=32).
// Heavy op: per-edge NNConv message = (x_src ⊗ hid_e) @ T2, T2[h*32+i, o] =
// nn.W2[h, i*32+o], computed fully fused with f32 WMMA (16x16x4), never
// materializing the per-edge [32,32] weight matrices.
//
// Input index map (setup_inputs() insertion order, params flattened
// depth-first in insertion order):
//   0 var_node_features [NV,2] f32   1 con_node_features [NC,2] f32
//   2 node_types [N] i32             3 edge_index [2,E] i32
//   4 edge_types [E,2] f32           5 assoc_var [NV] i32
//   6 assoc_con [NC] i32
//   7..10  var_mlp  W1(2,32) b1(32) W2(32,32) b2(32)
//   11..14 con_mlp  W1 b1 W2 b2
//   15+6L..20+6L conv{L+1}: nn.W1(2,32) nn.b1(32) nn.W2(32,1024) nn.b2(1024)
//                           root(32,32) bias(32)     for L=0..3
//   39,40 fc1 W(160,32) b(32)   41,42 fc2 W(32,32) b(32)
//   43,44 fc3 W(32,32) b(32)    45,46 fc6 W(32,2)  b(2)
// Output: log_softmax logits [NV,2] f32.
// ---------------------------------------------------------------------------

typedef float v2f __attribute__((ext_vector_type(2)));
typedef float v8f __attribute__((ext_vector_type(8)));

static __device__ inline v8f wmma_f32_k4(v2f a, v2f b, v8f c) {
  // D(16x16,f32) = A(16x4,f32) * B(4x16,f32) + C
  return __builtin_amdgcn_wmma_f32_16x16x4_f32(false, a, false, b,
                                               (short)0, c, false, false);
}

// ---------------------------------------------------------------------------
__global__ __launch_bounds__(256)
void zero_kernel(float* __restrict__ p, int n) {
  int i = blockIdx.x * blockDim.x + threadIdx.x;
  int stride = gridDim.x * blockDim.x;
  for (; i < n; i += stride) p[i] = 0.0f;
}

// ---------------------------------------------------------------------------
// Node embedding: x[slot] = mlp2(features)  for var and con nodes.
__global__ __launch_bounds__(256)
void embed_kernel(const float* __restrict__ vf, const float* __restrict__ cf,
                  const int* __restrict__ avar, const int* __restrict__ acon,
                  const float* __restrict__ vW1, const float* __restrict__ vb1,
                  const float* __restrict__ vW2, const float* __restrict__ vb2,
                  const float* __restrict__ cW1, const float* __restrict__ cb1,
                  const float* __restrict__ cW2, const float* __restrict__ cb2,
                  float* __restrict__ x, int NVn, int NCn) {
  __shared__ float sW1[2][64];
  __shared__ float sb1[2][32];
  __shared__ float sW2[2][1024];
  __shared__ float sb2[2][32];
  for (int i = threadIdx.x; i < 64; i += blockDim.x) { sW1[0][i] = vW1[i]; sW1[1][i] = cW1[i]; }
  for (int i = threadIdx.x; i < 32; i += blockDim.x) {
    sb1[0][i] = vb1[i]; sb1[1][i] = cb1[i];
    sb2[0][i] = vb2[i]; sb2[1][i] = cb2[i];
  }
  for (int i = threadIdx.x; i < 1024; i += blockDim.x) { sW2[0][i] = vW2[i]; sW2[1][i] = cW2[i]; }
  __syncthreads();

  int idx = blockIdx.x * blockDim.x + threadIdx.x;
  int total = NVn + NCn;
  if (idx >= total) return;
  int t = (idx < NVn) ? 0 : 1;
  int r = t ? (idx - NVn) : idx;
  const float* f = t ? cf : vf;
  float f0 = f[2 * r], f1 = f[2 * r + 1];
  float h[32];
#pragma unroll
  for (int j = 0; j < 32; ++j)
    h[j] = fmaxf(f0 * sW1[t][j] + f1 * sW1[t][32 + j] + sb1[t][j], 0.0f);
  int slot = t ? acon[r] : avar[r];
  for (int o = 0; o < 32; ++o) {
    float acc = sb2[t][o];
#pragma unroll
    for (int j = 0; j < 32; ++j) acc += h[j] * sW2[t][j * 32 + o];
    x[(size_t)slot * 32 + o] = acc;
  }
}

// ---------------------------------------------------------------------------
// Per-edge incoming-degree counts (layer-invariant).
__global__ __launch_bounds__(256)
void count_kernel(const int* __restrict__ eidx, float* __restrict__ cnt, int E) {
  int e = blockIdx.x * blockDim.x + threadIdx.x;
  if (e < E) atomicAdd(&cnt[eidx[E + e]], 1.0f);
}

// ---------------------------------------------------------------------------
// Fused NNConv edge kernel (WMMA f32 16x16x4):
//   hid = relu(edge_attr @ nnW1 + nnb1)                      (per edge, 32)
//   msg[e,o] = sum_{h,i} hid[e,h]*x[src,i]*W2[h, i*32+o]
//            + sum_i     x[src,i]*b2[i*32+o]
//   atomicAdd msg into sum[dst,:]
//
// One wave owns 4 M-tiles (64 edges). The Kronecker K loop is hh(32)*kb(8);
// B operands (4 global dwords, L2/WGP$-resident W2 slices) are shared across
// the 4 M-tiles, x_src slices live in registers (xreg), hid is re-read from
// LDS once per hh. Inner step: 4 vmem + 4 v_pk_mul + 8 v_wmma.
#define TPW 4  // M-tiles per wave
__global__ __launch_bounds__(128)
void conv_edge_kernel(const float* __restrict__ x,      // [N,32]
                      const int* __restrict__ eidx,     // [2,E]
                      const float* __restrict__ etype,  // [E,2]
                      const float* __restrict__ nnW1,   // [2,32]
                      const float* __restrict__ nnb1,   // [32]
                      const float* __restrict__ nnW2,   // [32,1024]
                      const float* __restrict__ nnb2,   // [1024]
                      float* __restrict__ sum,          // [N,32]
                      int E, int nGroups) {
  // 4 waves/block, 64 edges/wave: 2 * 4 * 64*32 floats = 64 KB
  __shared__ __align__(16) float lds_hid[4][64 * 32];
  __shared__ __align__(16) float lds_xs[4][64 * 32];

  const int lane = threadIdx.x & 31;
  const int wave = threadIdx.x >> 5;
  float* hidT = lds_hid[wave];
  float* xsT = lds_xs[wave];

  const int groupsPerIter = gridDim.x * 4;
  const int nIter = (nGroups + groupsPerIter - 1) / groupsPerIter;

  for (int iter = 0; iter < nIter; ++iter) {
    const int group = (iter * gridDim.x + blockIdx.x) * 4 + wave;
    const bool valid = (group < nGroups);
    const int base = group * (16 * TPW);

    if (valid) {
      // Stage per-edge hidden vector (edge MLP layer 1) and gathered x_src.
      const float w1a = nnW1[lane];
      const float w1b = nnW1[32 + lane];
      const float b1v = nnb1[lane];
      for (int it = 0; it < 16 * TPW; ++it) {
        int e = base + it;
        float h = 0.0f, xv = 0.0f;
        if (e < E) {
          float t0 = etype[2 * e], t1 = etype[2 * e + 1];
          h = fmaxf(t0 * w1a + t1 * w1b + b1v, 0.0f);
          int src = eidx[e];
          xv = x[(size_t)src * 32 + lane];
        }
        hidT[it * 32 + lane] = h;
        xsT[it * 32 + lane] = xv;
      }
    }
    __syncthreads();

    if (valid) {
      const int row = lane & 15;   // A: M index; B/C/D: N index
      const int half = lane >> 4;  // 0/1
      const int koff = half * 2;   // K sub-index within each 4-block
      const int n = lane & 15;
      const v2f* xsT2 = (const v2f*)xsT;

      // Per-lane x_src slice registers: xreg[t][kb] covers K = 4kb+koff(+1).
      v2f xreg[TPW][8];
#pragma unroll
      for (int t = 0; t < TPW; ++t) {
#pragma unroll
        for (int kb = 0; kb < 8; ++kb)
          xreg[t][kb] = xsT2[(t * 16 + row) * 16 + kb * 2 + half];
      }

      v8f acc[TPW][2];
#pragma unroll
      for (int t = 0; t < TPW; ++t) {
        acc[t][0] = (v8f){0, 0, 0, 0, 0, 0, 0, 0};
        acc[t][1] = (v8f){0, 0, 0, 0, 0, 0, 0, 0};
      }

      for (int hh = 0; hh < 32; ++hh) {
        float hv[TPW];
#pragma unroll
        for (int t = 0; t < TPW; ++t) hv[t] = hidT[(t * 16 + row) * 32 + hh];
#pragma unroll
        for (int kb = 0; kb < 8; ++kb) {
          const float* wbase = nnW2 + hh * 1024 + (kb * 4 + koff) * 32;
          v2f b0 = {wbase[n], wbase[32 + n]};       // o in [0,16)
          v2f b1 = {wbase[16 + n], wbase[48 + n]};  // o in [16,32)
#pragma unroll
          for (int t = 0; t < TPW; ++t) {
            v2f a;
            a.x = hv[t] * xreg[t][kb].x;
            a.y = hv[t] * xreg[t][kb].y;
            acc[t][0] = wmma_f32_k4(a, b0, acc[t][0]);
            acc[t][1] = wmma_f32_k4(a, b1, acc[t][1]);
          }
        }
      }

      // bias part: msg += x_src @ B2, B2[i,o] = nnb2[i*32+o]  (K=32)
#pragma unroll
      for (int kb = 0; kb < 8; ++kb) {
        const float* bb = nnb2 + (kb * 4 + koff) * 32;
        v2f b0 = {bb[n], bb[32 + n]};
        v2f b1 = {bb[16 + n], bb[48 + n]};
#pragma unroll
        for (int t = 0; t < TPW; ++t) {
          acc[t][0] = wmma_f32_k4(xreg[t][kb], b0, acc[t][0]);
          acc[t][1] = wmma_f32_k4(xreg[t][kb], b1, acc[t][1]);
        }
      }

      // Scatter: C/D layout -> lane holds (M = v + 8*half, N = lane&15).
#pragma unroll
      for (int t = 0; t < TPW; ++t) {
#pragma unroll
        for (int v = 0; v < 8; ++v) {
          int M = v + 8 * half;
          int e = base + t * 16 + M;
          if (e < E) {
            int dst = eidx[E + e];
            atomicAdd(&sum[(size_t)dst * 32 + n], acc[t][0][v]);
            atomicAdd(&sum[(size_t)dst * 32 + 16 + n], acc[t][1][v]);
          }
        }
      }
    }
    __syncthreads();
  }
}

// ---------------------------------------------------------------------------
// x_out = relu(sum/max(cnt,1) + x_in @ root + bias)
__global__ __launch_bounds__(256)
void node_update_kernel(const float* __restrict__ xin,
                        const float* __restrict__ sum,
                        const float* __restrict__ cnt,
                        const float* __restrict__ root,
                        const float* __restrict__ bias,
                        float* __restrict__ xout, int Nn) {
  __shared__ float sroot[1024];
  __shared__ float sbias[32];
  for (int i = threadIdx.x; i < 1024; i += blockDim.x) sroot[i] = root[i];
  if (threadIdx.x < 32) sbias[threadIdx.x] = bias[threadIdx.x];
  __syncthreads();
  int idx = blockIdx.x * blockDim.x + threadIdx.x;
  if (idx >= Nn * 32) return;
  int v = idx >> 5, o = idx & 31;
  float acc = sbias[o];
#pragma unroll
  for (int i = 0; i < 32; ++i) acc += xin[(size_t)v * 32 + i] * sroot[i * 32 + o];
  float c = cnt[v];
  c = (c > 1.0f) ? c : 1.0f;
  acc += sum[idx] / c;
  xout[idx] = fmaxf(acc, 0.0f);
}

// ---------------------------------------------------------------------------
// Head: h = concat(x0..x4)[assoc_var]; fc1..fc3 relu; fc6; log_softmax.
// One wave per var node; lane = output unit.
__global__ __launch_bounds__(128)
void head_kernel(const float* __restrict__ x0, const float* __restrict__ x1,
                 const float* __restrict__ x2, const float* __restrict__ x3,
                 const float* __restrict__ x4, const int* __restrict__ avar,
                 const float* __restrict__ W1, const float* __restrict__ b1,
                 const float* __restrict__ W2, const float* __restrict__ b2,
                 const float* __restrict__ W3, const float* __restrict__ b3,
                 const float* __restrict__ W6, const float* __restrict__ b6,
                 float* __restrict__ out, int NVn) {
  __shared__ float hbuf[4][160];
  __shared__ float habuf[4][32];
  const int lane = threadIdx.x & 31;
  const int wave = threadIdx.x >> 5;
  int node = blockIdx.x * 4 + wave;
  const bool valid = (node < NVn);
  int nv = valid ? node : 0;
  int slot = avar[nv];
  float* H = hbuf[wave];
  float* HA = habuf[wave];

  H[lane] = x0[(size_t)slot * 32 + lane];
  H[32 + lane] = x1[(size_t)slot * 32 + lane];
  H[64 + lane] = x2[(size_t)slot * 32 + lane];
  H[96 + lane] = x3[(size_t)slot * 32 + lane];
  H[128 + lane] = x4[(size_t)slot * 32 + lane];
  __syncthreads();

  float a1 = b1[lane];
  for (int i = 0; i < 160; ++i) a1 += H[i] * W1[i * 32 + lane];
  a1 = fmaxf(a1, 0.0f);
  HA[lane] = a1;
  __syncthreads();

  float a2 = b2[lane];
#pragma unroll
  for (int i = 0; i < 32; ++i) a2 += HA[i] * W2[i * 32 + lane];
  a2 = fmaxf(a2, 0.0f);
  __syncthreads();
  HA[lane] = a2;
  __syncthreads();

  float a3 = b3[lane];
#pragma unroll
  for (int i = 0; i < 32; ++i) a3 += HA[i] * W3[i * 32 + lane];
  a3 = fmaxf(a3, 0.0f);

  float p0 = a3 * W6[lane * 2 + 0];
  float p1 = a3 * W6[lane * 2 + 1];
#pragma unroll
  for (int off = 16; off > 0; off >>= 1) {
    p0 += __shfl_xor(p0, off, 32);
    p1 += __shfl_xor(p1, off, 32);
  }
  float l0 = p0 + b6[0];
  float l1 = p1 + b6[1];
  float m = fmaxf(l0, l1);
  float lse = m + logf(expf(l0 - m) + expf(l1 - m));
  if (valid && lane == 0) {
    out[(size_t)node * 2 + 0] = l0 - lse;
    out[(size_t)node * 2 + 1] = l1 - lse;
  }
}

// ---------------------------------------------------------------------------
extern "C" void kernel_launch(void* const* d_in, const int* in_sizes, int n_in,
                              void* d_out, int out_size, void* d_ws,
                              size_t ws_size, hipStream_t stream) {
  const float* vf = (const float*)d_in[0];
  const float* cf = (const float*)d_in[1];
  const int* eidx = (const int*)d_in[3];
  const float* etype = (const float*)d_in[4];
  const int* avar = (const int*)d_in[5];
  const int* acon = (const int*)d_in[6];

  const int NVn = in_sizes[0] / 2;
  const int NCn = in_sizes[1] / 2;
  const int Nn = in_sizes[2];
  const int En = in_sizes[3] / 2;

  auto P = [&](int i) { return (const float*)d_in[i]; };

  // Workspace layout (f32): xs0..xs4 [N,32] each, sum [N,32], cnt [N].
  float* ws = (float*)d_ws;
  const size_t xsz = (size_t)Nn * 32;
  float* xbuf[5];
  for (int c = 0; c < 5; ++c) xbuf[c] = ws + c * xsz;
  float* msum = ws + 5 * xsz;
  float* cntf = ws + 6 * xsz;

  const int B = 256;
  zero_kernel<<<(int)((xsz + B - 1) / B), B, 0, stream>>>(xbuf[0], (int)xsz);
  zero_kernel<<<(Nn + B - 1) / B, B, 0, stream>>>(cntf, Nn);

  embed_kernel<<<(Nn + B - 1) / B, B, 0, stream>>>(
      vf, cf, avar, acon, P(7), P(8), P(9), P(10), P(11), P(12), P(13), P(14),
      xbuf[0], NVn, NCn);

  count_kernel<<<(En + B - 1) / B, B, 0, stream>>>(eidx, cntf, En);

  const int nGroups = (En + 16 * TPW - 1) / (16 * TPW);
  int convBlocks = (nGroups + 3) / 4;
  if (convBlocks > 8192) convBlocks = 8192;

  for (int L = 0; L < 4; ++L) {
    const int pb = 15 + 6 * L;  // conv{L+1}: nn.W1 nn.b1 nn.W2 nn.b2 root bias
    zero_kernel<<<(int)((xsz + B - 1) / B), B, 0, stream>>>(msum, (int)xsz);
    conv_edge_kernel<<<convBlocks, 128, 0, stream>>>(
        xbuf[L], eidx, etype, P(pb), P(pb + 1), P(pb + 2), P(pb + 3), msum, En,
        nGroups);
    node_update_kernel<<<(int)((xsz + B - 1) / B), B, 0, stream>>>(
        xbuf[L], msum, cntf, P(pb + 4), P(pb + 5), xbuf[L + 1], Nn);
  }

  head_kernel<<<(NVn + 3) / 4, 128, 0, stream>>>(
      xbuf[0], xbuf[1], xbuf[2], xbuf[3], xbuf[4], avar, P(39), P(40), P(41),
      P(42), P(43), P(44), P(45), P(46), (float*)d_out, NVn);
}